// simple_GCN_model_77309411328114
// MI455X (gfx1250) — compile-verified
//
#include <hip/hip_runtime.h>
#include <hip/hip_bf16.h>

#define GCN_N   50000
#define GCN_E   1600000
#define GCN_IN  256
#define GCN_H   64
#define GCN_C   16
#define GCN_L   2
#define GCN_NT  5000
#define GCN_EPS 1e-12f

typedef __attribute__((ext_vector_type(2))) float v2f;
typedef __attribute__((ext_vector_type(8))) float v8f;

// ---------------------------------------------------------------- utilities
__global__ void k_fill_f32(float* __restrict__ p, float v, int n) {
  int t = blockIdx.x * blockDim.x + threadIdx.x;
  if (t < n) p[t] = v;
}

__global__ void k_deg_accum(const int* __restrict__ dst, float* __restrict__ deg, int E) {
  int t = blockIdx.x * blockDim.x + threadIdx.x;
  if (t < E) atomicAdd(&deg[dst[t]], 1.0f);
}

__global__ void k_rsqrt_inplace(float* __restrict__ p, int n) {
  int t = blockIdx.x * blockDim.x + threadIdx.x;
  if (t < n) p[t] = rsqrtf(p[t]);
}

__global__ void k_relu_inplace(float* __restrict__ p, long n) {
  long t = (long)blockIdx.x * blockDim.x + threadIdx.x;
  if (t < n) p[t] = fmaxf(p[t], 0.0f);
}

// -------------------------------------------------------------------------
// fp32 GEMM via V_WMMA_F32_16X16X4_F32, Nn fixed at 64.
// One wave computes a 16x64 output block (4 accumulators) so each A-fragment
// load (global_load_b64) feeds 4 WMMAs.
// Fused epilogue: XW = A@B (needed raw by the edge scatter) and
//                 Hout = bias + dinv[row]^2 * XW  (the self-loop term).
// Layouts (ISA 7.12.2):
//   A 16x4 : lanes 0-15 row M=lane, K=0,1 in v0,v1; lanes 16-31 K=2,3.
//   B 4x16 : lanes 0-15 col N=lane, K=0,1 in v0,v1; lanes 16-31 K=2,3.
//   C 16x16: VGPR r -> row r (lanes 0-15) / row r+8 (lanes 16-31), col=lane&15.
// -------------------------------------------------------------------------
__global__ void k_gemm_wmma_f32_n64(const float* __restrict__ A, const float* __restrict__ B,
                                    float* __restrict__ XW, const float* __restrict__ dinv,
                                    const float* __restrict__ bias, float* __restrict__ Hout,
                                    int M, int K) {
  const int wave = (int)((blockIdx.x * blockDim.x + threadIdx.x) >> 5);
  const int lane = threadIdx.x & 31;           // wave32 on gfx1250
  const int tm   = wave;                       // 16-row block index
  if (tm * 16 >= M) return;                    // wave-uniform: EXEC all-1s for WMMA
  const int half = lane >> 4;                  // 0 -> K=0,1 ; 1 -> K=2,3
  const int l15  = lane & 15;
  const long arow = (long)(tm * 16 + l15);

  v8f acc[4] = {};
  const float* ap = A + arow * (long)K + half * 2;
  const float* bp = B + (long)(half * 2) * GCN_H + l15;   // col base for n-tile 0
  for (int k = 0; k < K; k += 4) {
    v2f a;
    a.x = ap[0];
    a.y = ap[1];
#pragma unroll
    for (int t = 0; t < 4; ++t) {
      v2f b;
      b.x = bp[t * 16];
      b.y = bp[GCN_H + t * 16];
      acc[t] = __builtin_amdgcn_wmma_f32_16x16x4_f32(
          /*neg_a=*/false, a, /*neg_b=*/false, b,
          /*c_mod=*/(short)0, acc[t], /*reuse_a=*/false, /*reuse_b=*/false);
    }
    ap += 4;
    bp += 4 * GCN_H;
  }

  float bcol[4];
#pragma unroll
  for (int t = 0; t < 4; ++t) bcol[t] = bias[t * 16 + l15];

  const long crow = (long)(tm * 16 + half * 8);
#pragma unroll
  for (int r = 0; r < 8; ++r) {
    const long row = crow + r;
    const float di = dinv[row];                // uniform per lane half (row r vs r+8)
    const float dsq = di * di;
    float* __restrict__ xwrow = XW + row * GCN_H;
    float* __restrict__ horow = Hout + row * GCN_H;
#pragma unroll
    for (int t = 0; t < 4; ++t) {
      const float v = acc[t][r];
      xwrow[t * 16 + l15] = v;
      horow[t * 16 + l15] = bcol[t] + dsq * v;
    }
  }
}

// ------------------------------------------------ edge gather/scatter (64 lanes/edge)
__global__ void k_edge_scatter(const int* __restrict__ src, const int* __restrict__ dst,
                               const float* __restrict__ dinv, const float* __restrict__ XW,
                               float* __restrict__ Hout) {
  long t = (long)blockIdx.x * blockDim.x + threadIdx.x;
  long e = t >> 6;
  int  f = (int)(t & 63);
  if (e >= (long)GCN_E) return;
  int s = src[e], d = dst[e];
  float w = dinv[s] * dinv[d];
  atomicAdd(&Hout[(long)d * GCN_H + f], w * XW[(long)s * GCN_H + f]);
}

// ------------------------------------------------ row L2-normalize + head + log_softmax
__global__ void k_norm_head(const float* __restrict__ Hin, const float* __restrict__ W3,
                            const float* __restrict__ b3, float* __restrict__ hout,
                            float* __restrict__ pred) {
  int i = blockIdx.x * blockDim.x + threadIdx.x;
  if (i >= GCN_N) return;
  const float* row = Hin + (long)i * GCN_H;
  float h[GCN_H];
  float s = 0.0f;
#pragma unroll
  for (int f = 0; f < GCN_H; ++f) { h[f] = row[f]; s += h[f] * h[f]; }
  const float inv = 1.0f / fmaxf(sqrtf(s), GCN_EPS);
  float* __restrict__ orow = hout + (long)i * GCN_H;
#pragma unroll
  for (int f = 0; f < GCN_H; ++f) { h[f] *= inv; orow[f] = h[f]; }

  float lg[GCN_C];
  float m = -3.0e38f;
#pragma unroll
  for (int c = 0; c < GCN_C; ++c) {
    float a = b3[c];
#pragma unroll
    for (int f = 0; f < GCN_H; ++f) a += h[f] * W3[f * GCN_C + c];
    lg[c] = a;
    m = fmaxf(m, a);
  }
  float se = 0.0f;
#pragma unroll
  for (int c = 0; c < GCN_C; ++c) se += expf(lg[c] - m);
  const float lse = logf(se);
  float* __restrict__ prow = pred + (long)i * GCN_C;
#pragma unroll
  for (int c = 0; c < GCN_C; ++c) prow[c] = lg[c] - m - lse;
}

// ------------------------------------------------ NLL loss over train indices
__global__ void k_loss_zero(float* acc) { acc[0] = 0.0f; }

__global__ void k_loss_accum(const int* __restrict__ tidx, const int* __restrict__ labels,
                             const float* __restrict__ pred, float* __restrict__ acc) {
  int t = blockIdx.x * blockDim.x + threadIdx.x;
  if (t >= GCN_NT) return;
  int i = tidx[t];
  int l = labels[i];
  atomicAdd(acc, -pred[(long)i * GCN_C + l]);
}

__global__ void k_loss_final(const float* __restrict__ acc, float* __restrict__ out0) {
  out0[0] = acc[0] / (float)GCN_NT;
}

// ---------------------------------------------------------------- launcher
extern "C" void kernel_launch(void* const* d_in, const int* in_sizes, int n_in,
                              void* d_out, int out_size, void* d_ws, size_t ws_size,
                              hipStream_t stream) {
  const float* feats = (const float*)d_in[0];
  const float* W1    = (const float*)d_in[1];
  const float* b1    = (const float*)d_in[2];
  const float* W2    = (const float*)d_in[3];   // [L,H,H]
  const float* b2    = (const float*)d_in[4];   // [L,H]
  const float* W3    = (const float*)d_in[5];
  const float* b3    = (const float*)d_in[6];
  const int*   edge  = (const int*)d_in[7];     // [2,E]
  const int*   labs  = (const int*)d_in[8];
  const int*   tidx  = (const int*)d_in[9];
  const int* esrc = edge;
  const int* edst = edge + GCN_E;

  float* ws   = (float*)d_ws;
  float* dinv = ws;                                    // N
  float* XW   = dinv + GCN_N;                          // N*H
  float* HA   = XW + (size_t)GCN_N * GCN_H;            // N*H
  float* HB   = HA + (size_t)GCN_N * GCN_H;            // N*H
  float* lacc = HB + (size_t)GCN_N * GCN_H;            // 1

  float* out      = (float*)d_out;
  float* loss_out = out;                               // [1]
  float* pred     = out + 1;                           // [N,C]
  float* hout     = out + 1 + (size_t)GCN_N * GCN_C;   // [N,H]

  const dim3 B(256);
  auto grid = [](long n) { return dim3((unsigned)((n + 255) / 256)); };

  // degrees (self-loop => init 1), then dinv = rsqrt(deg)
  k_fill_f32<<<grid(GCN_N), B, 0, stream>>>(dinv, 1.0f, GCN_N);
  k_deg_accum<<<grid(GCN_E), B, 0, stream>>>(edst, dinv, GCN_E);
  k_rsqrt_inplace<<<grid(GCN_N), B, 0, stream>>>(dinv, GCN_N);

  const long nh = (long)GCN_N * GCN_H;
  const long scatter_threads = (long)GCN_E * GCN_H;
  const long gemm_threads = ((long)GCN_N / 16) * 32;   // one wave per 16x64 block

  // conv1: XW = feats @ W1 ; HA = b1 + dinv^2*XW ; scatter edges (no relu)
  k_gemm_wmma_f32_n64<<<grid(gemm_threads), B, 0, stream>>>(
      feats, W1, XW, dinv, b1, HA, GCN_N, GCN_IN);
  k_edge_scatter<<<grid(scatter_threads), B, 0, stream>>>(esrc, edst, dinv, XW, HA);

  // hidden convs: relu(aggregate(H @ W2[l]) + b2[l])
  float* cur = HA;
  float* nxt = HB;
  for (int l = 0; l < GCN_L; ++l) {
    k_gemm_wmma_f32_n64<<<grid(gemm_threads), B, 0, stream>>>(
        cur, W2 + (size_t)l * GCN_H * GCN_H, XW, dinv, b2 + l * GCN_H, nxt, GCN_N, GCN_H);
    k_edge_scatter<<<grid(scatter_threads), B, 0, stream>>>(esrc, edst, dinv, XW, nxt);
    k_relu_inplace<<<grid(nh), B, 0, stream>>>(nxt, nh);
    float* tmp = cur; cur = nxt; nxt = tmp;
  }

  // normalize rows -> h output, classifier head -> prediction (fused), then loss
  k_norm_head<<<grid(GCN_N), B, 0, stream>>>(cur, W3, b3, hout, pred);
  k_loss_zero<<<1, 1, 0, stream>>>(lacc);
  k_loss_accum<<<grid(GCN_NT), B, 0, stream>>>(tidx, labs, pred, lacc);
  k_loss_final<<<1, 1, 0, stream>>>(lacc, loss_out);
}